// GCNlayer_19164144075537
// MI455X (gfx1250) — compile-verified
//
#include <hip/hip_runtime.h>
#include <hip/hip_bf16.h>

typedef __attribute__((ext_vector_type(2))) float v2f;
typedef __attribute__((ext_vector_type(4))) float v4f;
typedef __attribute__((ext_vector_type(8))) float v8f;

#define LAMDA   0.1f
#define EPSV    1e-14f
#define FDIM    128
#define KN      32
#define HDIM    3
#define ODIM    100
#define ROWS    16          // batch rows per block
#define WAVES   16
#define HF_STR  132         // padded Hfeat row stride (dwords)

__global__ __launch_bounds__(512) void gcn_fused(
    const float* __restrict__ x,      // [B,128]
    const float* __restrict__ Nx,     // [B,32,128]
    const float* __restrict__ mu,     // [1,3]
    const float* __restrict__ sigma,  // [1,3]
    const float* __restrict__ Aw,     // [128,3]
    const float* __restrict__ Abias,  // [3]
    const float* __restrict__ theta,  // [128,100]
    const float* __restrict__ bias,   // [100]
    float* __restrict__ out,          // [B,100]
    int B)
{
    __shared__ __attribute__((aligned(16))) float s_Aw[FDIM * HDIM];   // 384 floats
    __shared__ float s_w[WAVES][KN];                                   // per-row neighbor weights
    __shared__ __attribute__((aligned(16))) float s_hf[ROWS][HF_STR];  // Hfeat tile (padded)

    const int tid  = threadIdx.x;
    const int wave = tid >> 5;
    const int lane = tid & 31;

    if (tid < FDIM * HDIM) s_Aw[tid] = Aw[tid];
    __syncthreads();

    const int b = blockIdx.x * ROWS + wave;
    if (b < B) {
        // ---------- Phase A1: lane = neighbor k; single HBM pass over Nx row ----------
        const float* nxrow = Nx + ((size_t)b * KN + lane) * FDIM;
        const float* xrow  = x + (size_t)b * FDIM;
        float a0 = 0.f, a1 = 0.f, a2 = 0.f;
        #pragma unroll 4
        for (int f = 0; f < FDIM; f += 4) {
            v4f nv = *(const v4f*)(nxrow + f);
            v4f xv = *(const v4f*)(xrow + f);      // broadcast within wave
            v4f d  = xv - nv;
            v4f w0 = *(const v4f*)(s_Aw + f * 3);       // A[f][0..2],  A[f+1][0]
            v4f w1 = *(const v4f*)(s_Aw + f * 3 + 4);   // A[f+1][1..2],A[f+2][0..1]
            v4f w2 = *(const v4f*)(s_Aw + f * 3 + 8);   // A[f+2][2],   A[f+3][0..2]
            a0 = fmaf(d.x, w0.x, a0); a1 = fmaf(d.x, w0.y, a1); a2 = fmaf(d.x, w0.z, a2);
            a0 = fmaf(d.y, w0.w, a0); a1 = fmaf(d.y, w1.x, a1); a2 = fmaf(d.y, w1.y, a2);
            a0 = fmaf(d.z, w1.z, a0); a1 = fmaf(d.z, w1.w, a1); a2 = fmaf(d.z, w2.x, a2);
            a0 = fmaf(d.w, w2.y, a0); a1 = fmaf(d.w, w2.z, a1); a2 = fmaf(d.w, w2.w, a2);
        }
        float m0 = tanhf(a0 + Abias[0]);
        float m1 = tanhf(a1 + Abias[1]);
        float m2 = tanhf(a2 + Abias[2]);
        float s0 = sigma[0], s1 = sigma[1], s2 = sigma[2];
        float u0 = mu[0],    u1 = mu[1],    u2 = mu[2];
        float dm = -0.5f * ((m0 - u0) * (m0 - u0) / (EPSV + s0 * s0)
                          + (m1 - u1) * (m1 - u1) / (EPSV + s1 * s1)
                          + (m2 - u2) * (m2 - u2) / (EPSV + s2 * s2));
        float e = __expf(dm);

        // wave32 reduction over all 32 neighbors
        float S = e;
        #pragma unroll
        for (int off = 16; off > 0; off >>= 1) S += __shfl_xor(S, off, 32);

        float wn     = LAMDA * e / (EPSV + S);
        float sumw   = LAMDA * S / (EPSV + S);
        float invDen = 1.0f / (1.0f + sumw);
        s_w[wave][lane] = wn;   // same-wave LDS ops are in order; no barrier needed

        // ---------- Phase A2: lane = feature chunk; Nx re-read served by L2/WGP$ ----------
        const float* nxbase = Nx + (size_t)b * KN * FDIM;
        const int fo = lane * 4;
        v4f xm = {0.f, 0.f, 0.f, 0.f};
        #pragma unroll 8
        for (int k = 0; k < KN; ++k) {
            v4f nv   = *(const v4f*)(nxbase + k * FDIM + fo);
            float wk = s_w[wave][k];    // broadcast read
            xm += nv * wk;
        }
        v4f xv = *(const v4f*)(xrow + fo);
        v4f hf = (xm + xv) * invDen;
        *(v4f*)(&s_hf[wave][fo]) = hf;
    }
    __syncthreads();

    // ---------- Phase B: Hfeat[16,128] @ theta[128,100] via V_WMMA_F32_16X16X4_F32 ----------
    if (wave < 7) {                         // 7 tiles of 16 cols cover O=100
        const int hl   = lane >> 4;         // half-wave selects K pair (0,1) vs (2,3)
        const int nloc = lane & 15;
        const int col  = wave * 16 + nloc;
        const int colc = (col < ODIM) ? col : (ODIM - 1);

        v8f c = {};
        #pragma unroll 8
        for (int kk = 0; kk < FDIM; kk += 4) {
            const int k0 = kk + 2 * hl;
            // A frag: lane holds row M=nloc, K elements k0,k0+1 (ISA 16x4 f32 layout)
            v2f afrag = *(const v2f*)(&s_hf[nloc][k0]);
            // B frag: lane holds col N, K elements k0,k0+1
            v2f bfrag;
            bfrag.x = theta[k0 * ODIM + colc];
            bfrag.y = theta[(k0 + 1) * ODIM + colc];
            c = __builtin_amdgcn_wmma_f32_16x16x4_f32(
                    false, afrag, false, bfrag, (short)0, c, false, false);
        }

        const float bv    = bias[colc];
        const int   baseb = blockIdx.x * ROWS;
        if (col < ODIM) {
            // D layout: VGPR j -> row M = j + 8*hl, col N = lane%16 (+tile offset)
            float* orow = out + (size_t)(baseb + 8 * hl) * ODIM + col;
            if (baseb + ROWS <= B) {
                // common path: full 16-row tile, straight-line stores with imm offsets
                #pragma unroll
                for (int j = 0; j < 8; ++j) {
                    float v = c[j] + bv;
                    orow[(size_t)j * ODIM] = v > 0.f ? v : 0.f;
                }
            } else {
                #pragma unroll
                for (int j = 0; j < 8; ++j) {
                    if (baseb + 8 * hl + j < B) {
                        float v = c[j] + bv;
                        orow[(size_t)j * ODIM] = v > 0.f ? v : 0.f;
                    }
                }
            }
        }
    }
}

extern "C" void kernel_launch(void* const* d_in, const int* in_sizes, int n_in,
                              void* d_out, int out_size, void* d_ws, size_t ws_size,
                              hipStream_t stream) {
    const float* x     = (const float*)d_in[0];
    const float* Nx    = (const float*)d_in[1];
    const float* mu    = (const float*)d_in[2];
    const float* sigma = (const float*)d_in[3];
    const float* Aw    = (const float*)d_in[4];
    const float* Abias = (const float*)d_in[5];
    const float* theta = (const float*)d_in[6];
    const float* bias  = (const float*)d_in[7];
    float* out = (float*)d_out;

    const int B = in_sizes[0] / FDIM;
    const int grid = (B + ROWS - 1) / ROWS;
    gcn_fused<<<grid, 512, 0, stream>>>(x, Nx, mu, sigma, Aw, Abias, theta, bias, out, B);
}